// SpectralConsistencyLoss_65738769433119
// MI455X (gfx1250) — compile-verified
//
#include <hip/hip_runtime.h>
#include <math.h>

// Problem constants (from reference setup_inputs)
#define B_  2
#define C_  48
#define H_  80
#define W_  80
#define D_  48
#define K_  2
#define N_  (H_ * W_ * D_)   // 307200 voxels per batch
#define WD_ (W_ * D_)        // 3840
#define CHUNK_ 256           // voxels per block; N_/CHUNK_ = 1200 exactly
#define NCHUNK_ (N_ / CHUNK_)

// Workspace layout (floats)
#define WS_CSUM 0     // [B*K*C] = 192 : per-class channel sums
#define WS_CNT  192   // [B*K]   = 4   : per-class voxel counts
#define WS_CEN  196   // [B*K*C] = 192 : finalized centers
#define WS_ACC  388   // [7] : 0..3 = class dist sums (b*2+k), 4=lh, 5=lw, 6=ld
#define WS_ZERO 512   // floats to clear

typedef float v2f __attribute__((ext_vector_type(2)));
typedef float v8f __attribute__((ext_vector_type(8)));

// ---------------------------------------------------------------- zero ws
__global__ void k_zero(float* ws) {
    int i = blockIdx.x * blockDim.x + threadIdx.x;
    if (i < WS_ZERO) ws[i] = 0.0f;
}

// ---------------------------------------------------------------- counts n[b][k]
__global__ void k_count(const int* __restrict__ tgt, float* __restrict__ ws) {
    float c[4] = {0.f, 0.f, 0.f, 0.f};
    for (int i = blockIdx.x * blockDim.x + threadIdx.x; i < B_ * N_;
         i += gridDim.x * blockDim.x) {
        int b = i / N_;
        int t = tgt[i] & 1;
        c[b * 2 + t] += 1.0f;
    }
    __shared__ float s[4];
    if (threadIdx.x < 4) s[threadIdx.x] = 0.f;
    __syncthreads();
    #pragma unroll
    for (int j = 0; j < 4; ++j) atomicAdd(&s[j], c[j]);
    __syncthreads();
    if (threadIdx.x < 4) atomicAdd(&ws[WS_CNT + threadIdx.x], s[threadIdx.x]);
}

// ---------------------------------------------------------------- centers via WMMA
// center_sum[b][k][c] = sum_n onehot(n,k) * feat[b][c][n]
// A (16x4 f32): rows = 16 channels, cols = 4 voxels.
//   ISA layout: lanes 0-15 hold M=lane, K=0(x),1(y); lanes 16-31 hold M=lane-16, K=2(x),3(y).
// B (4x16 f32): rows = 4 voxels, cols = 16 classes (only 0,1 nonzero).
//   lanes 0-15: rows K=0(x),1(y), col=lane; lanes 16-31: rows K=2(x),3(y), col=lane-16.
// D (16x16 f32, 8 VGPRs): lane<16 -> D[v][lane], lane>=16 -> D[v+8][lane-16].
__global__ void __launch_bounds__(96) k_centers(const float* __restrict__ feat,
                                                const int* __restrict__ tgt,
                                                float* __restrict__ ws) {
    __shared__ float sF[C_ * 257];   // padded stride (257) to avoid bank conflicts
    __shared__ int   sT[CHUNK_];

    const int tid   = threadIdx.x;
    const int b     = blockIdx.x / NCHUNK_;
    const int nbase = (blockIdx.x % NCHUNK_) * CHUNK_;

    // Coalesced stage of a 48 x 256 feature tile (fastest axis = voxel).
    for (int e = tid; e < C_ * CHUNK_; e += 96) {
        int c = e >> 8;
        int v = e & 255;
        sF[c * 257 + v] = feat[(size_t)(b * C_ + c) * N_ + nbase + v];
    }
    for (int v = tid; v < CHUNK_; v += 96)
        sT[v] = tgt[(size_t)b * N_ + nbase + v] & 1;
    __syncthreads();

    const int wave = tid >> 5;          // 3 waves -> 3 channel tiles of 16
    const int lane = tid & 31;
    const int row  = lane & 15;
    const int hi   = lane >> 4;         // 0: K=0,1 ; 1: K=2,3
    const int col  = row;               // B/D column index for this lane
    const int crow = wave * 16 + row;   // channel for A rows

    v8f acc = {0.f, 0.f, 0.f, 0.f, 0.f, 0.f, 0.f, 0.f};
    for (int kk = 0; kk < CHUNK_ / 4; ++kk) {
        const int va = kk * 4 + hi * 2;
        v2f a, bv;
        a.x  = sF[crow * 257 + va];
        a.y  = sF[crow * 257 + va + 1];
        bv.x = (sT[va]     == col) ? 1.0f : 0.0f;
        bv.y = (sT[va + 1] == col) ? 1.0f : 0.0f;
        acc  = __builtin_amdgcn_wmma_f32_16x16x4_f32(
                   false, a, false, bv, (short)0, acc, false, false);
    }

    // Columns 0/1 of D hold the class sums.
    if (col < 2) {
        const int k     = col;
        const int mbase = wave * 16 + hi * 8;
        float* csum = ws + WS_CSUM + (b * K_ + k) * C_;
        #pragma unroll
        for (int v = 0; v < 8; ++v) atomicAdd(&csum[mbase + v], acc[v]);
    }
}

// ---------------------------------------------------------------- center = sum / max(n,1)
__global__ void k_fin_centers(float* __restrict__ ws) {
    int i = threadIdx.x;                 // 192 threads
    if (i < B_ * K_ * C_) {
        int bk = i / C_;
        ws[WS_CEN + i] = ws[WS_CSUM + i] / fmaxf(ws[WS_CNT + bk], 1.0f);
    }
}

// ---------------------------------------------------------------- fused per-voxel pass
__global__ void __launch_bounds__(256) k_main(const float* __restrict__ feat,
                                              const float* __restrict__ pred,
                                              const int* __restrict__ tgt,
                                              float* __restrict__ ws) {
    __shared__ float cen[K_ * C_];
    __shared__ float red[5];

    const int tid = threadIdx.x;
    const int b   = blockIdx.x / NCHUNK_;
    const int n   = (blockIdx.x % NCHUNK_) * CHUNK_ + tid;

    if (tid < K_ * C_) cen[tid] = ws[WS_CEN + b * K_ * C_ + tid];
    if (tid < 5) red[tid] = 0.f;
    __syncthreads();

    const int h   = n / WD_;
    const int rem = n % WD_;
    const int w   = rem / D_;
    const int d   = rem % D_;
    const int t   = tgt[(size_t)b * N_ + n] & 1;

    // softmax over K=2 at voxel n
    const float x0 = pred[(size_t)(b * K_ + 0) * N_ + n];
    const float x1 = pred[(size_t)(b * K_ + 1) * N_ + n];
    float m   = fmaxf(x0, x1);
    float e0  = expf(x0 - m), e1 = expf(x1 - m);
    float inv = 1.0f / (e0 + e1);
    float p0 = e0 * inv, p1 = e1 * inv;
    float conf = fmaxf(p0, p1);
    float pt   = (t == 0) ? p0 : p1;

    const bool okH = (h < H_ - 1);
    const bool okW = (w < W_ - 1);
    const bool okD = (d < D_ - 1);

    float d2 = 0.f, sh = 0.f, sw = 0.f, sd = 0.f;
    const size_t base = (size_t)b * C_ * N_ + n;
    for (int c = 0; c < C_; ++c) {
        const size_t idx = base + (size_t)c * N_;
        float f  = feat[idx];
        float df = f - cen[t * C_ + c];
        d2 += df * df;
        if (okH) { float v = feat[idx + WD_] - f; sh += v * v; }
        if (okW) { float v = feat[idx + D_]  - f; sw += v * v; }
        if (okD) { float v = feat[idx + 1]   - f; sd += v * v; }
    }

    float dist = sqrtf(d2 + 1e-12f);
    float cl0 = (t == 0) ? pt * dist : 0.f;
    float cl1 = (t == 1) ? pt * dist : 0.f;

    float lh = 0.f, lw = 0.f;
    if (okH) {
        float y0 = pred[(size_t)(b * K_ + 0) * N_ + n + WD_];
        float y1 = pred[(size_t)(b * K_ + 1) * N_ + n + WD_];
        float mm = fmaxf(y0, y1);
        float f0 = expf(y0 - mm), f1 = expf(y1 - mm);
        float cf = fmaxf(f0, f1) / (f0 + f1);
        lh = sh * 0.5f * (conf + cf);
    }
    if (okW) {
        float y0 = pred[(size_t)(b * K_ + 0) * N_ + n + D_];
        float y1 = pred[(size_t)(b * K_ + 1) * N_ + n + D_];
        float mm = fmaxf(y0, y1);
        float f0 = expf(y0 - mm), f1 = expf(y1 - mm);
        float cf = fmaxf(f0, f1) / (f0 + f1);
        lw = sw * 0.5f * (conf + cf);
    }

    atomicAdd(&red[0], cl0);
    atomicAdd(&red[1], cl1);
    atomicAdd(&red[2], lh);
    atomicAdd(&red[3], lw);
    atomicAdd(&red[4], sd);
    __syncthreads();
    if (tid < 2)       atomicAdd(&ws[WS_ACC + b * 2 + tid], red[tid]);
    else if (tid == 2) atomicAdd(&ws[WS_ACC + 4], red[2]);
    else if (tid == 3) atomicAdd(&ws[WS_ACC + 5], red[3]);
    else if (tid == 4) atomicAdd(&ws[WS_ACC + 6], red[4]);
}

// ---------------------------------------------------------------- final scalar combine
__global__ void k_final(const float* __restrict__ ws, float* __restrict__ out) {
    if (threadIdx.x != 0 || blockIdx.x != 0) return;

    float compact_sum = 0.f, vcount = 0.f;
    float pair_loss = 0.f, pair_count = 0.f;

    for (int b = 0; b < B_; ++b) {
        for (int k = 0; k < K_; ++k) {
            float nn = ws[WS_CNT + b * K_ + k];
            float cl = ws[WS_ACC + b * K_ + k] / fmaxf(nn, 1.0f);
            if (nn >= 2.0f) { compact_sum += cl; vcount += 1.0f; }
        }
        float ds = 0.f;
        for (int c = 0; c < C_; ++c) {
            float df = ws[WS_CEN + b * K_ * C_ + c] -
                       ws[WS_CEN + b * K_ * C_ + C_ + c];
            ds += df * df;
        }
        float dd = sqrtf(ds + 1e-12f);
        float l  = fmaxf(0.0f, 1.0f - dd);   // relu(MARGIN - d)
        bool present = (ws[WS_CNT + b * K_] > 0.f) && (ws[WS_CNT + b * K_ + 1] > 0.f);
        if (present) { pair_loss += l; pair_count += 1.0f; }
    }

    float compact = (vcount > 0.f) ? compact_sum / vcount : 0.f;
    float sep     = (pair_count > 0.f) ? pair_loss / pair_count : 0.f;

    float lh = ws[WS_ACC + 4] / (float)(B_ * (H_ - 1) * W_ * D_);
    float lw = ws[WS_ACC + 5] / (float)(B_ * H_ * (W_ - 1) * D_);
    float ld = ws[WS_ACC + 6] / (float)((long long)B_ * C_ * H_ * W_ * (D_ - 1)) * 0.1f;
    float smooth = lh + lw + ld;

    out[0] = 1.0f * compact + 0.5f * sep + 0.3f * smooth;
}

// ---------------------------------------------------------------- launch
extern "C" void kernel_launch(void* const* d_in, const int* in_sizes, int n_in,
                              void* d_out, int out_size, void* d_ws, size_t ws_size,
                              hipStream_t stream) {
    (void)in_sizes; (void)n_in; (void)out_size; (void)ws_size;
    const float* feat = (const float*)d_in[0];
    const float* pred = (const float*)d_in[1];
    const int*   tgt  = (const int*)d_in[2];
    float* ws  = (float*)d_ws;
    float* out = (float*)d_out;

    k_zero<<<2, 256, 0, stream>>>(ws);
    k_count<<<480, 256, 0, stream>>>(tgt, ws);
    k_centers<<<B_ * NCHUNK_, 96, 0, stream>>>(feat, tgt, ws);
    k_fin_centers<<<1, 192, 0, stream>>>(ws);
    k_main<<<B_ * NCHUNK_, 256, 0, stream>>>(feat, pred, tgt, ws);
    k_final<<<1, 1, 0, stream>>>(ws, out);
}